// Attention_42485816492411
// MI455X (gfx1250) — compile-verified
//
#include <hip/hip_runtime.h>
#include <math.h>

#define B_      64
#define T_      1000
#define ENC_HD  512
#define ATTN_H  1024
#define ATTN_D  128
#define N_FILT  32
#define KSIZE   31
#define PAD     15
#define TSPLIT  8
#define TCHUNK  (T_ / TSPLIT)   // 125

typedef __attribute__((ext_vector_type(2))) float v2f;
typedef __attribute__((ext_vector_type(8))) float v8f;

// -------------------------------------------------------------------------
// Kernel 1: q projection  q[b,d] = sum_h query[b,h] * Q_w[d,h]
// M=64, N=128, K=1024 via V_WMMA_F32_16X16X4_F32 (exact fp32).
// grid = 4 (M tiles), block = 256 (8 waves = 8 N tiles).
// -------------------------------------------------------------------------
__global__ __launch_bounds__(256) void qproj_kernel(const float* __restrict__ query,
                                                    const float* __restrict__ Qw,
                                                    float* __restrict__ qproj) {
    const int wave = threadIdx.x >> 5;
    const int lane = threadIdx.x & 31;
    const int half = lane >> 4;       // which 16-lane half
    const int l15  = lane & 15;
    const int kk   = half * 2;        // A/B: lanes 0-15 hold K=0,1; lanes 16-31 hold K=2,3
    const int m0   = blockIdx.x * 16;
    const int n0   = wave * 16;

    const float* arow = query + (size_t)(m0 + l15) * ATTN_H;  // A[m][k] = query[m0+m][k]
    const float* brow = Qw    + (size_t)(n0 + l15) * ATTN_H;  // B[k][n] = Q_w[n0+n][k]

    v8f acc = {};
    for (int k0 = 0; k0 < ATTN_H; k0 += 4) {
        v2f a, b;
        a.x = arow[k0 + kk];  a.y = arow[k0 + kk + 1];
        b.x = brow[k0 + kk];  b.y = brow[k0 + kk + 1];
        acc = __builtin_amdgcn_wmma_f32_16x16x4_f32(false, a, false, b,
                                                    (short)0, acc, false, false);
    }
    // C/D layout: VGPR r -> (M=r, N=lane) for lanes 0-15, (M=8+r, N=lane-16) for 16-31
    #pragma unroll
    for (int r = 0; r < 8; ++r) {
        int row = m0 + r + half * 8;
        int col = n0 + l15;
        qproj[row * ATTN_D + col] = acc[r];
    }
}

// -------------------------------------------------------------------------
// Kernel 2: fused location-conv + L projection (WMMA) + tanh energy.
// One block per (b, 16-row t tile). 256 threads = 8 waves = 8 d-tiles of 16.
// Streams processed_encoder_output exactly once (33 MB pass).
// -------------------------------------------------------------------------
__global__ __launch_bounds__(256) void energy_kernel(const float* __restrict__ prev_attn,
                                                     const float* __restrict__ conv_w,
                                                     const float* __restrict__ Lw,
                                                     const float* __restrict__ Ww,
                                                     const float* __restrict__ pe,
                                                     const float* __restrict__ qproj,
                                                     float* __restrict__ energies) {
    __shared__ float sWin[2][46];           // conv input window (t0-15 .. t0+30)
    __shared__ float sConv[16][N_FILT];     // conv output tile (16 t x 32 filt)
    __shared__ float sQ[ATTN_D];
    __shared__ float sW[ATTN_D];
    __shared__ float sE[16];

    const int tid = threadIdx.x;
    const int b   = blockIdx.y;
    const int t0  = blockIdx.x * 16;

    if (tid < 92) {                          // 2 channels x 46 taps
        int i = tid / 46, j = tid % 46;
        int g = t0 - PAD + j;
        sWin[i][j] = (g >= 0 && g < T_) ? prev_attn[(b * 2 + i) * T_ + g] : 0.f;
    }
    if (tid < ATTN_D) {
        sQ[tid] = qproj[b * ATTN_D + tid];
        sW[tid] = Ww[tid];                   // W_w shape (1,128)
    }
    if (tid < 16) sE[tid] = 0.f;
    __syncthreads();

    // conv tile: conv[m][c] = sum_{i,k} win[i][m+k] * conv_w[c,i,k]
    #pragma unroll
    for (int e = tid; e < 16 * N_FILT; e += 256) {
        int m = e >> 5, c = e & 31;
        const float* w0 = conv_w + (c * 2) * KSIZE;
        float s = 0.f;
        #pragma unroll
        for (int k = 0; k < KSIZE; ++k)
            s = fmaf(sWin[0][m + k], w0[k], fmaf(sWin[1][m + k], w0[KSIZE + k], s));
        sConv[m][c] = s;
    }
    __syncthreads();

    // loc tile via WMMA: (16 t x 32 c) x (32 c x 16 d) per wave, K=32 in 8 steps
    const int wave = tid >> 5;
    const int lane = tid & 31;
    const int half = lane >> 4;
    const int l15  = lane & 15;
    const int kk   = half * 2;
    const int d0   = wave * 16;

    v8f acc = {};
    #pragma unroll
    for (int k0 = 0; k0 < N_FILT; k0 += 4) {
        v2f a, bb;
        a.x  = sConv[l15][k0 + kk];
        a.y  = sConv[l15][k0 + kk + 1];
        bb.x = Lw[(d0 + l15) * N_FILT + k0 + kk];       // B[k][n] = L_w[d0+n][k]
        bb.y = Lw[(d0 + l15) * N_FILT + k0 + kk + 1];
        acc = __builtin_amdgcn_wmma_f32_16x16x4_f32(false, a, false, bb,
                                                    (short)0, acc, false, false);
    }

    // energies: e[t] += sum_d tanh(q[d] + pe[b,t,d] + loc[t,d]) * W[d]
    const int   d  = d0 + l15;
    const float qd = sQ[d];
    const float wd = sW[d];
    #pragma unroll
    for (int r = 0; r < 8; ++r) {
        int Mr = r + half * 8;
        int t  = t0 + Mr;
        int tc = t < T_ ? t : T_ - 1;                      // clamp OOB loads
        float v = tanhf(qd + pe[(size_t)(b * T_ + tc) * ATTN_D + d] + acc[r]) * wd;
        v += __shfl_xor(v, 1, 32);                         // reduce over 16 lanes
        v += __shfl_xor(v, 2, 32);                         // (masks stay in half)
        v += __shfl_xor(v, 4, 32);
        v += __shfl_xor(v, 8, 32);
        if (l15 == 0 && t < T_) atomicAdd(&sE[Mr], v);     // ds_add_f32, 8 waves
    }
    __syncthreads();
    if (tid < 16) {
        int t = t0 + tid;
        if (t < T_) energies[b * T_ + t] = sE[tid];
    }
}

// -------------------------------------------------------------------------
// Kernel 3: softmax over T per batch row.
// -------------------------------------------------------------------------
__global__ __launch_bounds__(256) void softmax_kernel(const float* __restrict__ energies,
                                                      float* __restrict__ weights) {
    __shared__ float sred[256];
    __shared__ float sexp[T_];
    const int b = blockIdx.x;
    const int tid = threadIdx.x;

    float m = -1e30f;
    for (int t = tid; t < T_; t += 256) m = fmaxf(m, energies[b * T_ + t]);
    sred[tid] = m;
    __syncthreads();
    for (int s = 128; s > 0; s >>= 1) {
        if (tid < s) sred[tid] = fmaxf(sred[tid], sred[tid + s]);
        __syncthreads();
    }
    m = sred[0];
    __syncthreads();

    float sum = 0.f;
    for (int t = tid; t < T_; t += 256) {
        float e = expf(energies[b * T_ + t] - m);
        sexp[t] = e;
        sum += e;
    }
    sred[tid] = sum;
    __syncthreads();
    for (int s = 128; s > 0; s >>= 1) {
        if (tid < s) sred[tid] += sred[tid + s];
        __syncthreads();
    }
    float inv = 1.f / sred[0];
    for (int t = tid; t < T_; t += 256) weights[b * T_ + t] = sexp[t] * inv;
}

// -------------------------------------------------------------------------
// Kernel 4: context partials (the 131 MB streaming pass).
// grid (TSPLIT, B), 128 threads, float4 per thread -> fully coalesced b128.
// Deterministic (no float global atomics).
// -------------------------------------------------------------------------
__global__ __launch_bounds__(128) void ctx_partial_kernel(const float* __restrict__ enc,
                                                          const float* __restrict__ weights,
                                                          float* __restrict__ partials) {
    __shared__ float sw[TCHUNK];
    const int b    = blockIdx.y;
    const int s    = blockIdx.x;
    const int tid  = threadIdx.x;
    const int tbeg = s * TCHUNK;

    if (tid < TCHUNK) sw[tid] = weights[b * T_ + tbeg + tid];
    __syncthreads();

    const int h = tid * 4;
    const float* base = enc + (size_t)(b * T_ + tbeg) * ENC_HD + h;
    float4 acc = {0.f, 0.f, 0.f, 0.f};
    for (int t = 0; t < TCHUNK; ++t) {
        float4 e = *(const float4*)(base + (size_t)t * ENC_HD);
        float w = sw[t];
        acc.x = fmaf(w, e.x, acc.x);
        acc.y = fmaf(w, e.y, acc.y);
        acc.z = fmaf(w, e.z, acc.z);
        acc.w = fmaf(w, e.w, acc.w);
    }
    float* p = partials + (size_t)(b * TSPLIT + s) * ENC_HD + h;
    p[0] = acc.x; p[1] = acc.y; p[2] = acc.z; p[3] = acc.w;
}

// -------------------------------------------------------------------------
// Kernel 5: deterministic 8-way reduce into attn_context.
// -------------------------------------------------------------------------
__global__ __launch_bounds__(256) void ctx_reduce_kernel(const float* __restrict__ partials,
                                                         float* __restrict__ ctx) {
    int idx = blockIdx.x * 256 + threadIdx.x;   // 0 .. 64*512-1
    int b = idx >> 9;
    int h = idx & 511;
    float s = 0.f;
    #pragma unroll
    for (int p = 0; p < TSPLIT; ++p)
        s += partials[(size_t)(b * TSPLIT + p) * ENC_HD + h];
    ctx[idx] = s;
}

// -------------------------------------------------------------------------
extern "C" void kernel_launch(void* const* d_in, const int* in_sizes, int n_in,
                              void* d_out, int out_size, void* d_ws, size_t ws_size,
                              hipStream_t stream) {
    const float* query = (const float*)d_in[0];   // (64, 1024)
    const float* enc   = (const float*)d_in[1];   // (64, 1000, 512)
    const float* pe    = (const float*)d_in[2];   // (64, 1000, 128)
    const float* prev  = (const float*)d_in[3];   // (64, 2, 1000)
    const float* Qw    = (const float*)d_in[4];   // (128, 1024)
    const float* Ww    = (const float*)d_in[5];   // (1, 128)
    const float* Lw    = (const float*)d_in[6];   // (128, 32)
    const float* convw = (const float*)d_in[7];   // (32, 2, 31)

    float* ctx     = (float*)d_out;               // attn_context (64*512)
    float* weights = (float*)d_out + B_ * ENC_HD; // attn_weights (64*1000)

    float* ws       = (float*)d_ws;
    float* qproj    = ws;                         // 64*128
    float* energies = qproj + B_ * ATTN_D;        // 64*1000
    float* partials = energies + B_ * T_;         // 64*8*512

    qproj_kernel<<<dim3(B_ / 16), 256, 0, stream>>>(query, Qw, qproj);
    energy_kernel<<<dim3((T_ + 15) / 16, B_), 256, 0, stream>>>(prev, convw, Lw, Ww,
                                                                pe, qproj, energies);
    softmax_kernel<<<dim3(B_), 256, 0, stream>>>(energies, weights);
    ctx_partial_kernel<<<dim3(TSPLIT, B_), 128, 0, stream>>>(enc, weights, partials);
    ctx_reduce_kernel<<<dim3(B_ * ENC_HD / 256), 256, 0, stream>>>(partials, ctx);
}